// ReadWriteHeads_61297773249161
// MI455X (gfx1250) — compile-verified
//
#include <hip/hip_runtime.h>
#include <hip/hip_bf16.h>
#include <math.h>

// ---------------- problem constants ----------------
constexpr int B_ = 1024;
constexpr int N_ = 256;
constexpr int D_ = 64;
constexpr int H_ = 4;
constexpr int IN_ = 512;
constexpr int GATE_ = 471;   // H*D + H + D + 1 + D + D + 1 + 1 + H + 3*H
constexpr float EPS_ = 1e-8f;

// gate offsets inside co[b, :]
constexpr int OFF_KR    = 0;          // H*D = 256
constexpr int OFF_BETAR = 256;        // H = 4
constexpr int OFF_KW    = 260;        // D = 64
constexpr int OFF_BETAW = 324;        // 1
constexpr int OFF_ERASE = 325;        // 64
constexpr int OFF_ADD   = 389;        // 64
constexpr int OFF_GA    = 453;        // 1
constexpr int OFF_GW    = 454;        // 1
constexpr int OFF_F     = 455;        // 4
constexpr int OFF_PI    = 459;        // 12

typedef float v2f __attribute__((ext_vector_type(2)));
typedef float v4f __attribute__((ext_vector_type(4)));   // native vector for nontemporal builtins
typedef float v8f __attribute__((ext_vector_type(8)));

__device__ __forceinline__ float sigmf(float x)     { return 1.0f / (1.0f + __expf(-x)); }
__device__ __forceinline__ float softplusf(float x) { return (x > 20.0f) ? x : log1pf(__expf(x)); }

// deterministic block-wide reductions over 256 threads
__device__ __forceinline__ float block_sum(float v, float* red, int n) {
    __syncthreads();
    red[n] = v; __syncthreads();
    #pragma unroll
    for (int s = 128; s > 0; s >>= 1) { if (n < s) red[n] += red[n + s]; __syncthreads(); }
    float r = red[0]; __syncthreads();
    return r;
}
__device__ __forceinline__ float block_max(float v, float* red, int n) {
    __syncthreads();
    red[n] = v; __syncthreads();
    #pragma unroll
    for (int s = 128; s > 0; s >>= 1) { if (n < s) red[n] = fmaxf(red[n], red[n + s]); __syncthreads(); }
    float r = red[0]; __syncthreads();
    return r;
}

// ---------------------------------------------------------------------------
// Kernel 1: controller GEMM  co = x(1024x512) @ W(512x471) + b  via fp32 WMMA
// One wave = one 16x16 output tile, K stepped by 4 (V_WMMA_F32_16X16X4_F32).
// A layout (16x4 f32): lane m (0..15)/m+16: VGPR0=K(2*half+0), VGPR1=K(2*half+1)
// B layout (4x16 f32): lanes 0-15 hold K=0/1 (VGPR0/1), lanes 16-31 K=2/3
// C/D layout: VGPR v -> (M = v + 8*half, N = lane&15)
//
// Out-of-range columns are CLAMPED, not zero-guarded: a B lane only feeds its
// own output column, and columns >= GATE_ are never stored. This keeps EXEC
// all-ones through the entire WMMA loop (no saveexec/branch per load).
// ---------------------------------------------------------------------------
__global__ void k_gemm(const float* __restrict__ x, const float* __restrict__ W,
                       const float* __restrict__ bias, float* __restrict__ co)
{
    const int lane = threadIdx.x;       // 0..31 (wave32)
    const int half = lane >> 4;
    const int l16  = lane & 15;
    const int tN = blockIdx.x;          // 0..29  (30*16 = 480 >= 471)
    const int tM = blockIdx.y;          // 0..63

    const int col  = tN * 16 + l16;
    const int colc = (col < GATE_) ? col : (GATE_ - 1);   // clamp (dup column, unused)
    const int rowA = tM * 16 + l16;
    const float bv = bias[colc];

    v8f c = {bv, bv, bv, bv, bv, bv, bv, bv};   // C pre-loaded with bias row

    const float* xr  = x + (size_t)rowA * IN_ + 2 * half;
    const float* wc0 = W + (size_t)(2 * half) * GATE_ + colc;

    for (int k = 0; k < IN_; k += 4) {
        v2f a  = { xr[k], xr[k + 1] };
        v2f bm = { wc0[(size_t)k * GATE_], wc0[(size_t)(k + 1) * GATE_] };
        c = __builtin_amdgcn_wmma_f32_16x16x4_f32(false, a, false, bm,
                                                  (short)0, c, false, false);
    }

    if (col < GATE_) {
        #pragma unroll
        for (int v = 0; v < 8; v++) {
            int row = tM * 16 + v + 8 * half;
            co[(size_t)row * GATE_ + col] = c[v];
        }
    }
}

// ---------------------------------------------------------------------------
// Kernel 2: usage update + allocation weighting (bitonic argsort in LDS)
// one block per batch row, 256 threads (thread n = slot n)
// ---------------------------------------------------------------------------
__global__ void k_usage_alloc(const float* __restrict__ co,
                              const float* __restrict__ prev_usage,
                              const float* __restrict__ prev_rw,   // (B,H,N)
                              const float* __restrict__ prev_ww,   // (B,1,N)
                              float* __restrict__ out_usage,
                              float* __restrict__ ws_alloc)
{
    const int b = blockIdx.x, n = threadIdx.x;
    __shared__ float sval[N_];
    __shared__ int   sidx[N_];
    __shared__ float scan[N_];
    __shared__ float f_s[H_];

    const float* cb = co + (size_t)b * GATE_;
    if (n < H_) f_s[n] = sigmf(cb[OFF_F + n]);
    __syncthreads();

    float psi = 1.0f;
    #pragma unroll
    for (int h = 0; h < H_; h++)
        psi *= 1.0f - f_s[h] * prev_rw[((size_t)b * H_ + h) * N_ + n];
    const float regw = prev_ww[(size_t)b * N_ + n];           // 1 - prod(1-w) over 1 head
    const float u    = 0.005f + 0.995f * prev_usage[(size_t)b * N_ + n];
    const float usage = (u + (1.0f - u) * regw) * psi;
    out_usage[(size_t)b * N_ + n] = usage;

    sval[n] = usage; sidx[n] = n;
    __syncthreads();

    // bitonic sort ascending, stable via index tiebreak (matches argsort)
    for (int k = 2; k <= N_; k <<= 1) {
        for (int j = k >> 1; j > 0; j >>= 1) {
            int ixj = n ^ j;
            if (ixj > n) {
                bool ascend = ((n & k) == 0);
                float va = sval[n], vb = sval[ixj];
                int   ia = sidx[n], ib = sidx[ixj];
                bool gt = (va > vb) || (va == vb && ia > ib);
                bool doswap = ascend ? gt : !gt;
                if (doswap) { sval[n] = vb; sval[ixj] = va; sidx[n] = ib; sidx[ixj] = ia; }
            }
            __syncthreads();
        }
    }

    // exclusive cumulative product (Hillis-Steele inclusive, then shift)
    scan[n] = sval[n];
    __syncthreads();
    for (int off = 1; off < N_; off <<= 1) {
        float t = (n >= off) ? scan[n - off] : 1.0f;
        __syncthreads();
        scan[n] *= t;
        __syncthreads();
    }
    float dps = (n == 0) ? 1.0f : scan[n - 1];
    float sa  = (1.0f - sval[n]) * dps;
    ws_alloc[(size_t)b * N_ + sidx[n]] = sa;   // scatter = unsort
}

// ---------------------------------------------------------------------------
// Kernel 3: write-head addressing + memory update (fused)
// ---------------------------------------------------------------------------
__global__ void k_write(const float* __restrict__ co, const float* __restrict__ mem,
                        const float* __restrict__ alloc, const float* __restrict__ prevp,
                        float* __restrict__ o_ww, float* __restrict__ o_p,
                        float* __restrict__ o_mem)
{
    const int b = blockIdx.x, n = threadIdx.x;
    __shared__ float kn_s[D_], er_s[D_], ad_s[D_];
    __shared__ float red[N_];
    __shared__ float gates[3];  // betaw, ga, gw

    const float* cb = co + (size_t)b * GATE_;
    if (n < D_) {
        kn_s[n] = tanhf(cb[OFF_KW + n]);
        er_s[n] = sigmf(cb[OFF_ERASE + n]);
        ad_s[n] = tanhf(cb[OFF_ADD + n]);
    }
    if (n == 0) gates[0] = softplusf(cb[OFF_BETAW]);
    if (n == 1) gates[1] = sigmf(cb[OFF_GA]);
    if (n == 2) gates[2] = sigmf(cb[OFF_GW]);
    __syncthreads();

    float sq = (n < D_) ? kn_s[n] * kn_s[n] : 0.0f;
    float kss = block_sum(sq, red, n);
    if (n < D_) kn_s[n] *= 1.0f / (sqrtf(kss) + EPS_);
    __syncthreads();

    // memory row n stays in registers; streamed once -> non-temporal load
    v4f m[16];
    const v4f* mr = (const v4f*)(mem + ((size_t)b * N_ + n) * D_);
    float dot = 0.0f, nrm = 0.0f;
    #pragma unroll
    for (int q = 0; q < 16; q++) {
        v4f v = __builtin_nontemporal_load(mr + q);
        m[q] = v;
        dot += v.x * kn_s[4*q] + v.y * kn_s[4*q+1] + v.z * kn_s[4*q+2] + v.w * kn_s[4*q+3];
        nrm += v.x * v.x + v.y * v.y + v.z * v.z + v.w * v.w;
    }
    float s = gates[0] * dot / (sqrtf(nrm) + EPS_);

    float smax = block_max(s, red, n);
    float e = __expf(s - smax);
    float ssum = block_sum(e, red, n);
    float wc = e / ssum;

    float ww = gates[2] * (gates[1] * alloc[(size_t)b * N_ + n] + (1.0f - gates[1]) * wc);
    o_ww[(size_t)b * N_ + n] = ww;

    float sumw = block_sum(ww, red, n);
    o_p[(size_t)b * N_ + n] = (1.0f - sumw) * prevp[(size_t)b * N_ + n] + ww;

    // new_mem IS reused (k_read) -> regular-temporal store, stays in L2
    v4f* om = (v4f*)(o_mem + ((size_t)b * N_ + n) * D_);
    #pragma unroll
    for (int q = 0; q < 16; q++) {
        v4f v = m[q];
        v.x = v.x * (1.0f - ww * er_s[4*q+0]) + ww * ad_s[4*q+0];
        v.y = v.y * (1.0f - ww * er_s[4*q+1]) + ww * ad_s[4*q+1];
        v.z = v.z * (1.0f - ww * er_s[4*q+2]) + ww * ad_s[4*q+2];
        v.w = v.w * (1.0f - ww * er_s[4*q+3]) + ww * ad_s[4*q+3];
        om[q] = v;
    }
}

// ---------------------------------------------------------------------------
// Kernel 4: fused L construction + fw/bw mat-vecs (prevL touched exactly once)
// thread t: fixed 4 columns j0..j0+3 (v4f, coalesced), loops over rows.
// fw reduced via wave32 shfl (all lanes of a wave share the same row here);
// bw accumulated in registers across rows, merged via fixed LDS slots.
// prevL / L are pure streams -> non-temporal + explicit prefetch ahead.
// ---------------------------------------------------------------------------
__global__ void k_link(const float* __restrict__ prevL, const float* __restrict__ prevp,
                       const float* __restrict__ prev_rw, const float* __restrict__ wwv,
                       float* __restrict__ o_L, float* __restrict__ ws_fw,
                       float* __restrict__ ws_bw)
{
    const int b = blockIdx.x, t = threadIdx.x;
    const int jq = t & 63;     // column quad index
    const int ig = t >> 6;     // row group 0..3
    const int j0 = jq * 4;
    const int lane = t & 31;

    __shared__ float ww_s[N_], p_s[N_];
    __shared__ float r_s[H_ * N_];
    __shared__ float fw_part[H_ * N_ * 2];   // (h,i,wavehalf)
    __shared__ float bw_part[H_ * N_ * 4];   // (h,j,ig)

    ww_s[t] = wwv[(size_t)b * N_ + t];
    p_s[t]  = prevp[(size_t)b * N_ + t];
    #pragma unroll
    for (int h = 0; h < H_; h++)
        r_s[h * N_ + t] = prev_rw[((size_t)b * H_ + h) * N_ + t];
    __syncthreads();

    const int w2 = (t >> 5) & 1;  // which of the two waves that share a row
    float bwacc[H_][4];
    #pragma unroll
    for (int h = 0; h < H_; h++)
        #pragma unroll
        for (int q = 0; q < 4; q++) bwacc[h][q] = 0.0f;

    const float* plbase = prevL + ((size_t)b * N_ + ig) * N_ + j0;

    for (int step = 0; step < 64; step++) {
        const int i = step * 4 + ig;
        const float wwi = ww_s[i];
        const v4f pl =
            __builtin_nontemporal_load((const v4f*)(plbase + (size_t)step * 4 * N_));

        // prefetch 8 row-steps ahead (uniform branch) -> global_prefetch_b8
        if (step + 8 < 64)
            __builtin_prefetch(plbase + (size_t)(step + 8) * 4 * N_, 0, 0);

        float Lq[4];
        Lq[0] = (i == j0 + 0) ? 0.0f : (1.0f - wwi - ww_s[j0+0]) * pl.x + wwi * p_s[j0+0];
        Lq[1] = (i == j0 + 1) ? 0.0f : (1.0f - wwi - ww_s[j0+1]) * pl.y + wwi * p_s[j0+1];
        Lq[2] = (i == j0 + 2) ? 0.0f : (1.0f - wwi - ww_s[j0+2]) * pl.z + wwi * p_s[j0+2];
        Lq[3] = (i == j0 + 3) ? 0.0f : (1.0f - wwi - ww_s[j0+3]) * pl.w + wwi * p_s[j0+3];

        v4f ov = { Lq[0], Lq[1], Lq[2], Lq[3] };
        __builtin_nontemporal_store(ov, (v4f*)(o_L + ((size_t)b * N_ + i) * N_ + j0));

        #pragma unroll
        for (int h = 0; h < H_; h++) {
            // fw[h][i] partial over this thread's 4 columns
            float fwp = Lq[0] * r_s[h * N_ + j0 + 0] + Lq[1] * r_s[h * N_ + j0 + 1]
                      + Lq[2] * r_s[h * N_ + j0 + 2] + Lq[3] * r_s[h * N_ + j0 + 3];
            #pragma unroll
            for (int msk = 16; msk > 0; msk >>= 1) fwp += __shfl_xor(fwp, msk, 32);
            if (lane == 0) fw_part[(h * N_ + i) * 2 + w2] = fwp;  // written exactly once

            // bw[h][j] contribution from row i (register accumulation)
            const float rhi = r_s[h * N_ + i];
            bwacc[h][0] += Lq[0] * rhi;
            bwacc[h][1] += Lq[1] * rhi;
            bwacc[h][2] += Lq[2] * rhi;
            bwacc[h][3] += Lq[3] * rhi;
        }
    }

    #pragma unroll
    for (int h = 0; h < H_; h++)
        #pragma unroll
        for (int q = 0; q < 4; q++)
            bw_part[(h * N_ + j0 + q) * 4 + ig] = bwacc[h][q];
    __syncthreads();

    #pragma unroll
    for (int h = 0; h < H_; h++) {
        ws_fw[((size_t)b * H_ + h) * N_ + t] =
            fw_part[(h * N_ + t) * 2 + 0] + fw_part[(h * N_ + t) * 2 + 1];
        ws_bw[((size_t)b * H_ + h) * N_ + t] =
            bw_part[(h * N_ + t) * 4 + 0] + bw_part[(h * N_ + t) * 4 + 1] +
            bw_part[(h * N_ + t) * 4 + 2] + bw_part[(h * N_ + t) * 4 + 3];
    }
}

// ---------------------------------------------------------------------------
// Kernel 5: read-head content addressing + pi-mix + read vectors
// ---------------------------------------------------------------------------
__global__ void k_read(const float* __restrict__ co, const float* __restrict__ nmem,
                       const float* __restrict__ fw, const float* __restrict__ bw,
                       float* __restrict__ o_rw, float* __restrict__ o_rv)
{
    const int b = blockIdx.x, n = threadIdx.x;
    __shared__ float kr_s[H_ * D_];
    __shared__ float red[N_];
    __shared__ float beta_s[H_], kninv_s[H_], pi_s[3 * H_];
    __shared__ float rw_s[H_ * N_];

    const float* cb = co + (size_t)b * GATE_;
    kr_s[n] = tanhf(cb[OFF_KR + n]);         // n covers all H*D = 256
    if (n < H_) {
        beta_s[n] = softplusf(cb[OFF_BETAR + n]);
        float a0 = cb[OFF_PI + n*3 + 0], a1 = cb[OFF_PI + n*3 + 1], a2 = cb[OFF_PI + n*3 + 2];
        float mx = fmaxf(a0, fmaxf(a1, a2));
        float e0 = __expf(a0 - mx), e1 = __expf(a1 - mx), e2 = __expf(a2 - mx);
        float inv = 1.0f / (e0 + e1 + e2);
        pi_s[n*3+0] = e0 * inv; pi_s[n*3+1] = e1 * inv; pi_s[n*3+2] = e2 * inv;
    }
    __syncthreads();

    red[n] = kr_s[n] * kr_s[n];
    __syncthreads();
    if (n < H_) {
        float s = 0.0f;
        for (int d = 0; d < D_; d++) s += red[n * D_ + d];
        kninv_s[n] = 1.0f / (sqrtf(s) + EPS_);
    }
    __syncthreads();
    kr_s[n] *= kninv_s[n >> 6];
    __syncthreads();

    // cosine similarities of new_mem row n vs all heads
    float dot[H_] = {0.0f, 0.0f, 0.0f, 0.0f};
    float nrm = 0.0f;
    const v4f* mr = (const v4f*)(nmem + ((size_t)b * N_ + n) * D_);
    #pragma unroll
    for (int q = 0; q < 16; q++) {
        v4f v = mr[q];
        nrm += v.x * v.x + v.y * v.y + v.z * v.z + v.w * v.w;
        #pragma unroll
        for (int h = 0; h < H_; h++) {
            dot[h] += v.x * kr_s[h*D_ + 4*q+0] + v.y * kr_s[h*D_ + 4*q+1]
                    + v.z * kr_s[h*D_ + 4*q+2] + v.w * kr_s[h*D_ + 4*q+3];
        }
    }
    const float minv = 1.0f / (sqrtf(nrm) + EPS_);

    for (int h = 0; h < H_; h++) {
        float s = beta_s[h] * dot[h] * minv;
        float mx = block_max(s, red, n);
        float e = __expf(s - mx);
        float sm = block_sum(e, red, n);
        float cr = e / sm;
        float rw = pi_s[h*3+0] * bw[((size_t)b * H_ + h) * N_ + n]
                 + pi_s[h*3+1] * cr
                 + pi_s[h*3+2] * fw[((size_t)b * H_ + h) * N_ + n];
        o_rw[((size_t)b * H_ + h) * N_ + n] = rw;
        rw_s[h * N_ + n] = rw;
    }
    __syncthreads();

    // rv[h][d] = sum_n rw[h][n] * new_mem[n][d]; thread -> (h,d), coalesced on d
    const int h = n >> 6, d = n & 63;
    float acc = 0.0f;
    for (int j = 0; j < N_; j++)
        acc += rw_s[h * N_ + j] * nmem[((size_t)b * N_ + j) * D_ + d];
    o_rv[((size_t)b * H_ + h) * D_ + d] = acc;
}

// ---------------------------------------------------------------------------
extern "C" void kernel_launch(void* const* d_in, const int* in_sizes, int n_in,
                              void* d_out, int out_size, void* d_ws, size_t ws_size,
                              hipStream_t stream)
{
    const float* x    = (const float*)d_in[0];  // ctrl_inputs (B,IN)
    const float* mem  = (const float*)d_in[1];  // memory_state (B,N,D)
    const float* pu   = (const float*)d_in[2];  // prev_usage_vector (B,N)
    const float* prw  = (const float*)d_in[3];  // prev_read_weights (B,H,N)
    const float* pww  = (const float*)d_in[4];  // prev_write_weights (B,1,N)
    const float* pL   = (const float*)d_in[5];  // prev_link_matrix (B,N,N)
    const float* pp   = (const float*)d_in[6];  // prev_precedence_weights (B,1,N)
    const float* W    = (const float*)d_in[7];  // (IN,GATE)
    const float* bias = (const float*)d_in[8];  // (GATE,)

    float* out   = (float*)d_out;
    float* o_rv  = out;                                  // (B,H,D)
    float* o_mem = o_rv  + (size_t)B_ * H_ * D_;         // (B,N,D)
    float* o_rw  = o_mem + (size_t)B_ * N_ * D_;         // (B,H,N)
    float* o_ww  = o_rw  + (size_t)B_ * H_ * N_;         // (B,1,N)
    float* o_us  = o_ww  + (size_t)B_ * N_;              // (B,N)
    float* o_p   = o_us  + (size_t)B_ * N_;              // (B,1,N)
    float* o_L   = o_p   + (size_t)B_ * N_;              // (B,N,N)

    float* ws    = (float*)d_ws;
    float* co    = ws;                                   // (B,GATE)
    float* alloc = co    + (size_t)B_ * GATE_;           // (B,N)
    float* fw    = alloc + (size_t)B_ * N_;              // (B,H,N)
    float* bw    = fw    + (size_t)B_ * H_ * N_;         // (B,H,N)

    k_gemm<<<dim3((GATE_ + 15) / 16, B_ / 16), dim3(32), 0, stream>>>(x, W, bias, co);
    k_usage_alloc<<<dim3(B_), dim3(N_), 0, stream>>>(co, pu, prw, pww, o_us, alloc);
    k_write<<<dim3(B_), dim3(N_), 0, stream>>>(co, mem, alloc, pp, o_ww, o_p, o_mem);
    k_link<<<dim3(B_), dim3(N_), 0, stream>>>(pL, pp, prw, o_ww, o_L, fw, bw);
    k_read<<<dim3(B_), dim3(N_), 0, stream>>>(co, o_mem, fw, bw, o_rw, o_rv);
}